// Attention_65317862637882
// MI455X (gfx1250) — compile-verified
//
#include <hip/hip_runtime.h>
#include <hip/hip_bf16.h>
#include <cmath>

typedef __attribute__((ext_vector_type(2))) float v2f;
typedef __attribute__((ext_vector_type(8))) float v8f;

#define BB 32
#define LL 2048
#define DD 1024
#define DD2 512
#define KK2 2048
#define CHUNK 128                 // L-rows per fused workgroup
#define NCHUNK (LL / CHUNK)       // 16 chunks per batch

// ---------------------------------------------------------------------------
// C[M,N] = A[M,K] @ Bt[N,K]^T  via V_WMMA_F32_16X16X4_F32 (exact fp32 WMMA).
// One wave (32 threads) per 16x16 output tile.
// ---------------------------------------------------------------------------
__global__ void wmma_gemm_qk(const float* __restrict__ A,
                             const float* __restrict__ Bt,
                             float* __restrict__ C, int N, int K) {
  const int lane   = threadIdx.x;      // 0..31
  const int m0     = blockIdx.x * 16;
  const int n0     = blockIdx.y * 16;
  const int half   = lane >> 4;        // 0 or 1
  const int l16    = lane & 15;
  const int kShift = half * 2;
  const float* aRow = A  + (size_t)(m0 + l16) * K + kShift;
  const float* bRow = Bt + (size_t)(n0 + l16) * K + kShift;
  v8f acc = {};
  for (int k0 = 0; k0 < K; k0 += 4) {
    v2f a = *(const v2f*)(aRow + k0);
    v2f b = *(const v2f*)(bRow + k0);
    acc = __builtin_amdgcn_wmma_f32_16x16x4_f32(false, a, false, b,
                                                (short)0, acc, false, false);
  }
  const int mBase = m0 + half * 8;
#pragma unroll
  for (int r = 0; r < 8; ++r)
    C[(size_t)(mBase + r) * N + (n0 + l16)] = acc[r];
}

// ---------------------------------------------------------------------------
// out[32,512] = tanh( [mixed | q][32,2048] @ Wout[512,2048]^T ), WMMA fp32.
// ---------------------------------------------------------------------------
__global__ void wmma_gemm_out(const float* __restrict__ mixed,
                              const float* __restrict__ q,
                              const float* __restrict__ Wout,
                              float* __restrict__ out) {
  const int lane   = threadIdx.x;
  const int m0     = blockIdx.x * 16;
  const int n0     = blockIdx.y * 16;
  const int half   = lane >> 4;
  const int l16    = lane & 15;
  const int kShift = half * 2;
  const float* mRow = mixed + (size_t)(m0 + l16) * DD;
  const float* qRow = q     + (size_t)(m0 + l16) * DD;
  const float* bRow = Wout  + (size_t)(n0 + l16) * KK2 + kShift;
  v8f acc = {};
  for (int k0 = 0; k0 < KK2; k0 += 4) {
    const int kk = k0 + kShift;
    const float* src = (kk < DD) ? (mRow + kk) : (qRow + (kk - DD));
    v2f a = *(const v2f*)src;
    v2f b = *(const v2f*)(bRow + k0);
    acc = __builtin_amdgcn_wmma_f32_16x16x4_f32(false, a, false, b,
                                                (short)0, acc, false, false);
  }
  const int mBase = m0 + half * 8;
#pragma unroll
  for (int r = 0; r < 8; ++r)
    out[(size_t)(mBase + r) * DD2 + (n0 + l16)] = tanhf(acc[r]);
}

// ---------------------------------------------------------------------------
// Fused per-(batch, L-chunk) kernel: single HBM stream over context.
//   Phase 1: scores for CHUNK rows (one wave per 4 rows), store to ws.
//   Phase 2: chunk-local softmax partials m_c, S_c; e_l, e_l*bt_l in LDS.
//   Phase 3: re-read same context slice (L2-hot) accumulating
//            P[d] = sum_l e_l * ( c + ae*bt_l*max(0,c) ).
// Block = 1024 threads (32 wave32s). Grid = (BB, NCHUNK).
// ---------------------------------------------------------------------------
__global__ void fused_chunk_kernel(const float* __restrict__ ctx,
                                   const float* __restrict__ q,
                                   const float* __restrict__ ae,
                                   const float* __restrict__ ab,
                                   float* __restrict__ scores,
                                   float* __restrict__ P,
                                   float* __restrict__ mc,
                                   float* __restrict__ Sc) {
  __shared__ float qs[DD];
  __shared__ float ls[CHUNK];
  __shared__ float es[CHUNK];
  __shared__ float ebs[CHUNK];
  __shared__ float red[CHUNK];

  const int b     = blockIdx.x;
  const int chunk = blockIdx.y;
  const int tid   = threadIdx.x;          // 0..1023
  const int l0    = chunk * CHUNK;
  const float aev = ae[b];
  const float abv = ab[b];

  // load q[b] into LDS
  qs[tid] = q[(size_t)b * DD + tid];
  __syncthreads();

  // ---- Phase 1: scores (wave per 4 rows) ----
  const int wave = tid >> 5;              // 0..31
  const int lane = tid & 31;
  const float4* q4 = (const float4*)qs;
#pragma unroll
  for (int i = 0; i < 4; ++i) {
    const int lr = wave * 4 + i;          // 0..127
    const float4* crow =
        (const float4*)(ctx + ((size_t)b * LL + l0 + lr) * DD);
    float acc = 0.f;
#pragma unroll
    for (int j = 0; j < 8; ++j) {
      const int idx = j * 32 + lane;
      float4 c = crow[idx];
      float4 w = q4[idx];
      acc += c.x * w.x + c.y * w.y + c.z * w.z + c.w * w.w;
    }
#pragma unroll
    for (int off = 16; off > 0; off >>= 1)
      acc += __shfl_xor(acc, off, 32);
    if (lane == 0) {
      ls[lr] = acc;
      scores[(size_t)b * LL + l0 + lr] = acc;
    }
  }
  __syncthreads();

  // ---- Phase 2: chunk-local max / exp / sum ----
  if (tid < CHUNK) red[tid] = ls[tid];
  __syncthreads();
  for (int s = CHUNK / 2; s > 0; s >>= 1) {
    if (tid < s) red[tid] = fmaxf(red[tid], red[tid + s]);
    __syncthreads();
  }
  const float m_c = red[0];
  __syncthreads();
  if (tid < CHUNK) {
    const float e = __expf(ls[tid] - m_c);
    es[tid]  = e;
    ebs[tid] = e * __expf(-abv * (float)(LL - 1 - (l0 + tid)));
    red[tid] = e;
  }
  __syncthreads();
  for (int s = CHUNK / 2; s > 0; s >>= 1) {
    if (tid < s) red[tid] += red[tid + s];
    __syncthreads();
  }
  if (tid == 0) {
    mc[b * NCHUNK + chunk] = m_c;
    Sc[b * NCHUNK + chunk] = red[0];
  }
  __syncthreads();

  // ---- Phase 3: partial mixed (context slice re-read, L2-hot) ----
  // thread tid owns column d = tid
  float acc = 0.f;
  const float* cbase = ctx + ((size_t)b * LL + l0) * DD + tid;
  for (int l = 0; l < CHUNK; ++l) {
    const float c  = cbase[(size_t)l * DD];
    const float e  = es[l];
    const float eb = ebs[l];
    acc += e * c + aev * eb * fmaxf(0.f, c);
  }
  P[(size_t)(b * NCHUNK + chunk) * DD + tid] = acc;
}

// ---------------------------------------------------------------------------
// Per batch: global softmax stats from chunk partials, rescale into mixed.
// ---------------------------------------------------------------------------
__global__ void combine_kernel(const float* __restrict__ P,
                               const float* __restrict__ mc,
                               const float* __restrict__ Sc,
                               float* __restrict__ mixed,
                               float* __restrict__ stats) {
  __shared__ float wv[NCHUNK];
  __shared__ float MZ[2];
  const int b   = blockIdx.x;
  const int tid = threadIdx.x;            // 0..255
  if (tid == 0) {
    float M = mc[b * NCHUNK];
    for (int c = 1; c < NCHUNK; ++c) M = fmaxf(M, mc[b * NCHUNK + c]);
    float Z = 0.f;
    for (int c = 0; c < NCHUNK; ++c) {
      const float w = __expf(mc[b * NCHUNK + c] - M);
      wv[c] = w;
      Z += w * Sc[b * NCHUNK + c];
    }
    MZ[0] = M;
    MZ[1] = 1.0f / Z;
    stats[b]      = M;
    stats[BB + b] = 1.0f / Z;
  }
  __syncthreads();
  const float invZ = MZ[1];
  for (int d = tid; d < DD; d += 256) {
    float s = 0.f;
#pragma unroll
    for (int c = 0; c < NCHUNK; ++c)
      s += wv[c] * P[(size_t)(b * NCHUNK + c) * DD + d];
    mixed[(size_t)b * DD + d] = s * invZ;
  }
}

// ---------------------------------------------------------------------------
// attn output: exp(score - M) / Z
// ---------------------------------------------------------------------------
__global__ void attn_kernel(const float* __restrict__ scores,
                            const float* __restrict__ stats,
                            float* __restrict__ attn) {
  const int b = blockIdx.x;
  const int l = blockIdx.y * 256 + threadIdx.x;
  attn[(size_t)b * LL + l] =
      __expf(scores[(size_t)b * LL + l] - stats[b]) * stats[BB + b];
}

// ---------------------------------------------------------------------------
extern "C" void kernel_launch(void* const* d_in, const int* in_sizes, int n_in,
                              void* d_out, int out_size, void* d_ws, size_t ws_size,
                              hipStream_t stream) {
  const float* query   = (const float*)d_in[0];  // [32,1,1024]
  const float* context = (const float*)d_in[1];  // [32,2048,1024]
  const float* W_in    = (const float*)d_in[2];  // [1024,1024]
  const float* W_out   = (const float*)d_in[3];  // [512,2048]
  const float* ae      = (const float*)d_in[4];  // [32]
  const float* ab      = (const float*)d_in[5];  // [32]

  float* out  = (float*)d_out;                   // [32,512]
  float* attn = out + BB * DD2;                  // [32,2048]

  float* ws     = (float*)d_ws;
  float* q      = ws;                            // 32768
  float* scores = q + BB * DD;                   // 65536
  float* P      = scores + BB * LL;              // 524288
  float* mc     = P + (size_t)BB * NCHUNK * DD;  // 512
  float* Sc     = mc + BB * NCHUNK;              // 512
  float* stats  = Sc + BB * NCHUNK;              // 64
  float* mixed  = stats + 2 * BB;                // 32768   (total ~2.6 MB)

  // 1) q = query @ W_in^T   (WMMA fp32)
  wmma_gemm_qk<<<dim3(BB / 16, DD / 16), 32, 0, stream>>>(query, W_in, q, DD, DD);
  // 2) fused scores + chunk softmax + chunk mixed partials (single context stream)
  fused_chunk_kernel<<<dim3(BB, NCHUNK), 1024, 0, stream>>>(context, q, ae, ab,
                                                            scores, P, mc, Sc);
  // 3) global softmax stats + mixed
  combine_kernel<<<BB, 256, 0, stream>>>(P, mc, Sc, mixed, stats);
  // 4) attn output
  attn_kernel<<<dim3(BB, LL / 256), 256, 0, stream>>>(scores, stats, attn);
  // 5) out = tanh([mixed|q] @ W_out^T)   (WMMA fp32)
  wmma_gemm_out<<<dim3(BB / 16, DD2 / 16), 32, 0, stream>>>(mixed, q, W_out, out);
}